// MlpMoeWithNoisyTopExpertsPerItemRouter_10316511445817
// MI455X (gfx1250) — compile-verified
//
#include <hip/hip_runtime.h>
#include <math.h>

// ---- problem constants (match reference) ----
#define E_    8
#define K_    2
#define D_    1024
#define M_    4096
#define GS_   1024
#define B_    8
#define S_    1024
#define G_    8          // (B*S)/GS
#define CAP_  256        // (K*GS)/E

typedef float v2f __attribute__((ext_vector_type(2)));
typedef float v8f __attribute__((ext_vector_type(8)));
typedef int   v4i __attribute__((vector_size(16)));   // matches builtin's param type

// ---- gfx1250 async global->LDS copy (ASYNCcnt-tracked), with safe fallback ----
#if __has_builtin(__builtin_amdgcn_global_load_async_to_lds_b128)
#define ASYNC_LDS 1
#else
#define ASYNC_LDS 0
#endif

__device__ __forceinline__ void async_cp16(const float* gsrc, float* ldst) {
#if ASYNC_LDS
    __builtin_amdgcn_global_load_async_to_lds_b128(
        (__attribute__((address_space(1))) v4i*)gsrc,
        (__attribute__((address_space(3))) v4i*)ldst, 0, 0);
#else
    (void)gsrc; (void)ldst;
#endif
}
__device__ __forceinline__ void wait_async0() {
#if ASYNC_LDS
    asm volatile("s_wait_asynccnt 0x0" ::: "memory");
#endif
}

__device__ __forceinline__ float gelu_tanh(float x) {
    const float kAlpha = 0.7978845608028654f;   // sqrt(2/pi)
    float x3 = x * x * x;
    return 0.5f * x * (1.0f + tanhf(kAlpha * (x + 0.044715f * x3)));
}

// -------------------- zero-init helpers --------------------
__global__ void moe_zero4(float4* p, int n) {
    int i = blockIdx.x * blockDim.x + threadIdx.x;
    if (i < n) p[i] = make_float4(0.f, 0.f, 0.f, 0.f);
}
__global__ void moe_zero1(float* p, int n) {
    int i = blockIdx.x * blockDim.x + threadIdx.x;
    if (i < n) p[i] = 0.f;
}

// -------------------- router: logits -> softmax -> top2 + imp --------------------
// one wave32 per token; 8 waves per 256-thread block
__global__ void moe_router(const float* __restrict__ x, const float* __restrict__ wr,
                           int* __restrict__ topi, float* __restrict__ topw,
                           float* __restrict__ imp) {
    int wave = threadIdx.x >> 5;
    int lane = threadIdx.x & 31;
    int t = blockIdx.x * 8 + wave;          // token row in [0, G*GS)
    int g = t >> 10;                        // / GS_
    int s = t & (GS_ - 1);
    const float* xrow = x + (size_t)t * D_;

    float acc[E_];
#pragma unroll
    for (int e = 0; e < E_; ++e) acc[e] = 0.f;

    for (int d = lane; d < D_; d += 32) {
        float xv = xrow[d];
        const float4* w4 = (const float4*)(wr + (size_t)d * E_);
        float4 wa = w4[0], wb = w4[1];
        acc[0] += xv * wa.x; acc[1] += xv * wa.y;
        acc[2] += xv * wa.z; acc[3] += xv * wa.w;
        acc[4] += xv * wb.x; acc[5] += xv * wb.y;
        acc[6] += xv * wb.z; acc[7] += xv * wb.w;
    }
#pragma unroll
    for (int e = 0; e < E_; ++e) {
        float v = acc[e];
#pragma unroll
        for (int off = 16; off > 0; off >>= 1) v += __shfl_xor(v, off, 32);
        acc[e] = v;
    }
    if (lane == 0) {
        float mx = acc[0];
#pragma unroll
        for (int e = 1; e < E_; ++e) mx = fmaxf(mx, acc[e]);
        float p[E_]; float sum = 0.f;
#pragma unroll
        for (int e = 0; e < E_; ++e) { p[e] = __expf(acc[e] - mx); sum += p[e]; }
        float inv = 1.f / sum;
#pragma unroll
        for (int e = 0; e < E_; ++e) p[e] *= inv;
        int e1 = 0;
#pragma unroll
        for (int e = 1; e < E_; ++e) if (p[e] > p[e1]) e1 = e;
        int e2 = (e1 == 0) ? 1 : 0;
#pragma unroll
        for (int e = 0; e < E_; ++e) if (e != e1 && p[e] > p[e2]) e2 = e;
        topi[(g * K_ + 0) * GS_ + s] = e1;  topw[(g * K_ + 0) * GS_ + s] = p[e1];
        topi[(g * K_ + 1) * GS_ + s] = e2;  topw[(g * K_ + 1) * GS_ + s] = p[e2];
#pragma unroll
        for (int e = 0; e < E_; ++e) atomicAdd(&imp[g * E_ + e], p[e]);
    }
}

// -------------------- k-major capacity scan (GShard priority order) --------------------
__global__ void moe_scan(const int* __restrict__ topi, const float* __restrict__ topw,
                         int* __restrict__ slot_token, float* __restrict__ slot_w) {
    int g = blockIdx.x;
    for (int i = threadIdx.x; i < E_ * CAP_; i += blockDim.x)
        slot_token[(size_t)g * E_ * CAP_ + i] = -1;
    __syncthreads();
    if (threadIdx.x == 0) {
        int cnt[E_];
#pragma unroll
        for (int e = 0; e < E_; ++e) cnt[e] = 0;
        for (int k = 0; k < K_; ++k) {
            for (int s = 0; s < GS_; ++s) {
                int idx = (g * K_ + k) * GS_ + s;
                int e = topi[idx];
                int p = cnt[e]++;
                if (p < CAP_) {
                    slot_token[((size_t)g * E_ + e) * CAP_ + p] = s;
                    slot_w[((size_t)g * E_ + e) * CAP_ + p] = topw[idx];
                }
            }
        }
    }
}

// -------------------- GEMM1: h = gelu(gather(x) @ wi + bi) --------------------
// block = 256 threads (8 waves). tile: 128 rows(C) x 64 cols(M). K-chunks of 32 over D.
// A tiles stream global->LDS via async DMA (double-buffered); B tiles are reg-staged and
// stored k-pair interleaved so each WMMA B fragment is one ds_load_b64.
__global__ void __launch_bounds__(256) moe_gemm1(const float* __restrict__ x,
                                                 const float* __restrict__ wi,
                                                 const float* __restrict__ bi,
                                                 const int* __restrict__ slot_token,
                                                 float* __restrict__ h) {
    __shared__ float Als[2][128][36];        // [buf][row][k] (+pad)
    __shared__ float Bls[2][16][128];        // [buf][k-pair][col*2 + j]

    int e = blockIdx.z / G_;
    int g = blockIdx.z % G_;
    int m0 = blockIdx.x * 64;
    int c0 = blockIdx.y * 128;
    int tid = threadIdx.x;
    int wave = tid >> 5, lane = tid & 31;

    // A staging: row + 16-float half; gather token once
    int ar = tid & 127;
    int ahalf = tid >> 7;
    int tok = slot_token[((size_t)g * E_ + e) * CAP_ + c0 + ar];
    const float* xrow = x + ((size_t)g * GS_ + (tok >= 0 ? tok : 0)) * D_;

    // B staging: k-pair bp in [0,16), column quad bq in [0,16)
    int bp = tid & 15;
    int bq = tid >> 4;
    const float* wie = wi + (size_t)e * D_ * M_;

    v8f acc[4] = {};
    int rowbase = wave * 16;
    int rrow = rowbase + (lane & 15);
    int kp = (lane >> 4) * 2;
    int cL = lane & 15;

    float4 bR0, bR1;
#if ASYNC_LDS
    // empty-slot rows: zero-fill both buffers once; async loads never touch them
    if (tok < 0) {
        float4 z = make_float4(0.f, 0.f, 0.f, 0.f);
#pragma unroll
        for (int b = 0; b < 2; ++b) {
            float4* d = (float4*)&Als[b][ar][ahalf * 16];
            d[0] = z; d[1] = z; d[2] = z; d[3] = z;
        }
    }
    auto stageA = [&](int kk, int buf) {     // async DMA global -> LDS, no VGPR staging
        if (tok >= 0) {
            const float* s = xrow + kk + ahalf * 16;
            float* d = &Als[buf][ar][ahalf * 16];
            async_cp16(s, d);         async_cp16(s + 4, d + 4);
            async_cp16(s + 8, d + 8); async_cp16(s + 12, d + 12);
        }
    };
#else
    float4 aR[4];
    auto loadA = [&](int kk) {
        if (tok >= 0) {
            const float4* s = (const float4*)(xrow + kk + ahalf * 16);
            aR[0] = s[0]; aR[1] = s[1]; aR[2] = s[2]; aR[3] = s[3];
        } else {
            float4 z = make_float4(0.f, 0.f, 0.f, 0.f);
            aR[0] = z; aR[1] = z; aR[2] = z; aR[3] = z;
        }
    };
    auto storeA = [&](int buf) {
        float4* d = (float4*)&Als[buf][ar][ahalf * 16];
        d[0] = aR[0]; d[1] = aR[1]; d[2] = aR[2]; d[3] = aR[3];
    };
#endif
    auto loadB = [&](int kk) {
        bR0 = *(const float4*)(wie + (size_t)(kk + 2 * bp)     * M_ + m0 + bq * 4);
        bR1 = *(const float4*)(wie + (size_t)(kk + 2 * bp + 1) * M_ + m0 + bq * 4);
    };
    auto storeB = [&](int buf) {
        float* base = &Bls[buf][bp][bq * 8];     // col = bq*4 -> col*2 = bq*8
        v2f t0 = {bR0.x, bR1.x}, t1 = {bR0.y, bR1.y};
        v2f t2 = {bR0.z, bR1.z}, t3 = {bR0.w, bR1.w};
        *(v2f*)(base + 0) = t0; *(v2f*)(base + 2) = t1;
        *(v2f*)(base + 4) = t2; *(v2f*)(base + 6) = t3;
    };

#if ASYNC_LDS
    stageA(0, 0); loadB(0); storeB(0); wait_async0();
#else
    loadA(0); loadB(0); storeA(0); storeB(0);
#endif
    __syncthreads();
    const int NCHUNK = D_ / 32;
    for (int i = 0; i < NCHUNK; ++i) {
        int buf = i & 1;
        if (i + 1 < NCHUNK) {                    // next chunk's loads fly during compute
#if ASYNC_LDS
            stageA((i + 1) * 32, buf ^ 1);
#else
            loadA((i + 1) * 32);
#endif
            loadB((i + 1) * 32);
            if (i + 2 < NCHUNK)
                __builtin_prefetch(wie + (size_t)((i + 2) * 32 + 2 * bp) * M_ + m0 + bq * 4, 0, 0);
        }
#pragma unroll
        for (int k4 = 0; k4 < 8; ++k4) {
            int kb = k4 * 4 + kp;                // even
            v2f a; a.x = Als[buf][rrow][kb]; a.y = Als[buf][rrow][kb + 1];
            int p = kb >> 1;
#pragma unroll
            for (int n = 0; n < 4; ++n) {
                v2f b = *(const v2f*)&Bls[buf][p][(n * 16 + cL) * 2];
                acc[n] = __builtin_amdgcn_wmma_f32_16x16x4_f32(
                    false, a, false, b, (short)0, acc[n], false, false);
            }
        }
        if (i + 1 < NCHUNK) {
#if !ASYNC_LDS
            storeA(buf ^ 1);
#endif
            storeB(buf ^ 1);
        }
        wait_async0();
        __syncthreads();
    }

    // epilogue: + bi, gelu, store h
    int roff = (lane >> 4) * 8;
#pragma unroll
    for (int n = 0; n < 4; ++n) {
        int mcol = m0 + n * 16 + cL;
        float bias = bi[(size_t)e * M_ + mcol];
#pragma unroll
        for (int v = 0; v < 8; ++v) {
            int c = c0 + rowbase + roff + v;
            float hv = gelu_tanh(acc[n][v] + bias);
            h[(((size_t)e * G_ + g) * CAP_ + c) * M_ + mcol] = hv;
        }
    }
}

// -------------------- GEMM2: ye = h @ wo + bo, fused combine-scatter --------------------
__global__ void __launch_bounds__(256) moe_gemm2(const float* __restrict__ h,
                                                 const float* __restrict__ wo,
                                                 const float* __restrict__ bo,
                                                 const int* __restrict__ slot_token,
                                                 const float* __restrict__ slot_w,
                                                 float* __restrict__ y) {
    __shared__ float Als[2][128][36];
    __shared__ float Bls[2][16][128];

    int e = blockIdx.z / G_;
    int g = blockIdx.z % G_;
    int d0 = blockIdx.x * 64;
    int c0 = blockIdx.y * 128;
    int tid = threadIdx.x;
    int wave = tid >> 5, lane = tid & 31;

    int ar = tid & 127;
    int ahalf = tid >> 7;
    const float* hrow = h + (((size_t)e * G_ + g) * CAP_ + c0 + ar) * M_;

    int bp = tid & 15;
    int bq = tid >> 4;
    const float* woe = wo + (size_t)e * M_ * D_;

    v8f acc[4] = {};
    int rowbase = wave * 16;
    int rrow = rowbase + (lane & 15);
    int kp = (lane >> 4) * 2;
    int cL = lane & 15;

    float4 bR0, bR1;
#if ASYNC_LDS
    auto stageA = [&](int kk, int buf) {
        const float* s = hrow + kk + ahalf * 16;
        float* d = &Als[buf][ar][ahalf * 16];
        async_cp16(s, d);         async_cp16(s + 4, d + 4);
        async_cp16(s + 8, d + 8); async_cp16(s + 12, d + 12);
    };
#else
    float4 aR[4];
    auto loadA = [&](int kk) {
        const float4* s = (const float4*)(hrow + kk + ahalf * 16);
        aR[0] = s[0]; aR[1] = s[1]; aR[2] = s[2]; aR[3] = s[3];
    };
    auto storeA = [&](int buf) {
        float4* d = (float4*)&Als[buf][ar][ahalf * 16];
        d[0] = aR[0]; d[1] = aR[1]; d[2] = aR[2]; d[3] = aR[3];
    };
#endif
    auto loadB = [&](int kk) {
        bR0 = *(const float4*)(woe + (size_t)(kk + 2 * bp)     * D_ + d0 + bq * 4);
        bR1 = *(const float4*)(woe + (size_t)(kk + 2 * bp + 1) * D_ + d0 + bq * 4);
    };
    auto storeB = [&](int buf) {
        float* base = &Bls[buf][bp][bq * 8];
        v2f t0 = {bR0.x, bR1.x}, t1 = {bR0.y, bR1.y};
        v2f t2 = {bR0.z, bR1.z}, t3 = {bR0.w, bR1.w};
        *(v2f*)(base + 0) = t0; *(v2f*)(base + 2) = t1;
        *(v2f*)(base + 4) = t2; *(v2f*)(base + 6) = t3;
    };

#if ASYNC_LDS
    stageA(0, 0); loadB(0); storeB(0); wait_async0();
#else
    loadA(0); loadB(0); storeA(0); storeB(0);
#endif
    __syncthreads();
    const int NCHUNK = M_ / 32;
    for (int i = 0; i < NCHUNK; ++i) {
        int buf = i & 1;
        if (i + 1 < NCHUNK) {
#if ASYNC_LDS
            stageA((i + 1) * 32, buf ^ 1);
#else
            loadA((i + 1) * 32);
#endif
            loadB((i + 1) * 32);
            if (i + 2 < NCHUNK)
                __builtin_prefetch(woe + (size_t)((i + 2) * 32 + 2 * bp) * D_ + d0 + bq * 4, 0, 0);
        }
#pragma unroll
        for (int k4 = 0; k4 < 8; ++k4) {
            int kb = k4 * 4 + kp;
            v2f a; a.x = Als[buf][rrow][kb]; a.y = Als[buf][rrow][kb + 1];
            int p = kb >> 1;
#pragma unroll
            for (int n = 0; n < 4; ++n) {
                v2f b = *(const v2f*)&Bls[buf][p][(n * 16 + cL) * 2];
                acc[n] = __builtin_amdgcn_wmma_f32_16x16x4_f32(
                    false, a, false, b, (short)0, acc[n], false, false);
            }
        }
        if (i + 1 < NCHUNK) {
#if !ASYNC_LDS
            storeA(buf ^ 1);
#endif
            storeB(buf ^ 1);
        }
        wait_async0();
        __syncthreads();
    }

    // epilogue: + bo, weight by gate, scatter-add into y (combine)
    int roff = (lane >> 4) * 8;
#pragma unroll
    for (int v = 0; v < 8; ++v) {
        int c = c0 + rowbase + roff + v;
        int tok = slot_token[((size_t)g * E_ + e) * CAP_ + c];
        if (tok < 0) continue;   // dropped/empty slot contributes nothing
        float wgt = slot_w[((size_t)g * E_ + e) * CAP_ + c];
        float* yrow = y + ((size_t)g * GS_ + tok) * D_;
#pragma unroll
        for (int n = 0; n < 4; ++n) {
            int dcol = d0 + n * 16 + cL;
            float val = acc[n][v] + bo[(size_t)e * D_ + dcol];
            atomicAdd(&yrow[dcol], wgt * val);
        }
    }
}

// -------------------- aux loss: mean over groups of (std/mean)^2 of imp --------------------
__global__ void moe_aux(const float* __restrict__ imp, float* __restrict__ out_aux) {
    int l = threadIdx.x;
    float cv2 = 0.f;
    if (l < G_) {
        float m = 0.f;
#pragma unroll
        for (int e = 0; e < E_; ++e) m += imp[l * E_ + e];
        m *= (1.f / E_);
        float var = 0.f;
#pragma unroll
        for (int e = 0; e < E_; ++e) { float d = imp[l * E_ + e] - m; var += d * d; }
        var *= (1.f / E_);
        cv2 = var / (m * m);
    }
#pragma unroll
    for (int off = 4; off > 0; off >>= 1) cv2 += __shfl_down(cv2, off, 32);
    if (l == 0) out_aux[0] = cv2 * (1.f / G_);
}

// -------------------- host launch --------------------
extern "C" void kernel_launch(void* const* d_in, const int* in_sizes, int n_in,
                              void* d_out, int out_size, void* d_ws, size_t ws_size,
                              hipStream_t stream) {
    const float* x  = (const float*)d_in[0];
    const float* wr = (const float*)d_in[1];
    const float* wi = (const float*)d_in[2];
    const float* bi = (const float*)d_in[3];
    const float* wo = (const float*)d_in[4];
    const float* bo = (const float*)d_in[5];
    float* y   = (float*)d_out;
    float* aux = y + (size_t)B_ * S_ * D_;

    char* ws = (char*)d_ws;
    size_t off = 0;
    auto take = [&](size_t bytes) { size_t o = off; off += (bytes + 255) & ~(size_t)255; return o; };
    int*   topi       = (int*)  (ws + take(sizeof(int)   * G_ * K_ * GS_));
    float* topw       = (float*)(ws + take(sizeof(float) * G_ * K_ * GS_));
    int*   slot_token = (int*)  (ws + take(sizeof(int)   * G_ * E_ * CAP_));
    float* slot_w     = (float*)(ws + take(sizeof(float) * G_ * E_ * CAP_));
    float* imp        = (float*)(ws + take(sizeof(float) * G_ * E_));
    float* h          = (float*)(ws + take(sizeof(float) * (size_t)E_ * G_ * CAP_ * M_));
    (void)in_sizes; (void)n_in; (void)out_size; (void)ws_size;

    // zero output (combine accumulates via atomics) and imp
    moe_zero4<<<(B_ * S_ * D_ / 4 + 255) / 256, 256, 0, stream>>>((float4*)y, B_ * S_ * D_ / 4);
    moe_zero1<<<1, 64, 0, stream>>>(imp, G_ * E_);

    moe_router<<<(G_ * GS_) / 8, 256, 0, stream>>>(x, wr, topi, topw, imp);
    moe_scan<<<G_, 64, 0, stream>>>(topi, topw, slot_token, slot_w);

    dim3 g1(M_ / 64, CAP_ / 128, E_ * G_);
    moe_gemm1<<<g1, 256, 0, stream>>>(x, wi, bi, slot_token, h);

    dim3 g2(D_ / 64, CAP_ / 128, E_ * G_);
    moe_gemm2<<<g2, 256, 0, stream>>>(h, wo, bo, slot_token, slot_w, y);

    moe_aux<<<1, 32, 0, stream>>>(imp, aux);
}